// DenoisingLocal_Global_ConvNN_2D_25039659335752
// MI455X (gfx1250) — compile-verified
//
#include <hip/hip_runtime.h>
#include <hip/hip_bf16.h>
#include <float.h>

typedef __attribute__((ext_vector_type(2))) float v2f;
typedef __attribute__((ext_vector_type(8))) float v8f;

#define B_  64
#define N_  1024
#define K_  9

// CDNA5 fp32 WMMA: D(16x16,f32) = A(16x4,f32) * B(4x16,f32) + C
__device__ __forceinline__ v8f wmma_f32_4(v2f a, v2f b, v8f c) {
  return __builtin_amdgcn_wmma_f32_16x16x4_f32(false, a, false, b, (short)0, c, false, false);
}

// ---------------------------------------------------------------------------
// Kernel 1: 3x3 conv (3->16, pad 1) + ReLU, written directly in token layout
// m1[(b*1024 + n)*64 + c], c = oc*4 + sy*2 + sx, n = h*32 + w (pixel-unshuffle fused)
// ---------------------------------------------------------------------------
__global__ void conv1_relu_tok(const float* __restrict__ x, const float* __restrict__ w,
                               const float* __restrict__ bias, float* __restrict__ m1) {
  int t = blockIdx.x * blockDim.x + threadIdx.x;
  if (t >= B_ * N_ * 64) return;
  int c = t & 63;
  int n = (t >> 6) & (N_ - 1);
  int b = t >> 16;
  int oc = c >> 2, sy = (c >> 1) & 1, sx = c & 1;
  int h = n >> 5, wc = n & 31;
  int Y = 2 * h + sy, X = 2 * wc + sx;
  float s = bias[oc];
  for (int ic = 0; ic < 3; ++ic)
    for (int dy = -1; dy <= 1; ++dy) {
      int yy = Y + dy; if ((unsigned)yy >= 64u) continue;
      for (int dx = -1; dx <= 1; ++dx) {
        int xx = X + dx; if ((unsigned)xx >= 64u) continue;
        s += x[((b * 3 + ic) * 64 + yy) * 64 + xx] *
             w[((oc * 3 + ic) * 3 + (dy + 1)) * 3 + (dx + 1)];
      }
    }
  m1[t] = fmaxf(s, 0.f);
}

// ---------------------------------------------------------------------------
// Kernel 2: per-token squared norm
// ---------------------------------------------------------------------------
template <int C>
__global__ void rownorm(const float* __restrict__ m, float* __restrict__ nsq) {
  int r = blockIdx.x * blockDim.x + threadIdx.x;
  if (r >= B_ * N_) return;
  const float* p = m + (size_t)r * C;
  float s = 0.f;
  for (int i = 0; i < C; ++i) s += p[i] * p[i];
  nsq[r] = s;
}

// ---------------------------------------------------------------------------
// Kernel 3: distance tiles via fp32 WMMA + per-row top-9 (stable, ascending j)
// One wave per 16-row strip; iterate 64 column tiles of 16.
// ---------------------------------------------------------------------------
template <int C>
__global__ __launch_bounds__(32) void topk_wmma(const float* __restrict__ m,
                                                const float* __restrict__ nsq,
                                                int* __restrict__ idxO) {
  __shared__ float As[16][C + 2];
  __shared__ float Bs[16][C + 2];
  __shared__ float Ds[16][17];
  __shared__ float nA[16], nB[16];

  int blk = blockIdx.x;
  int b = blk >> 6;
  int I = blk & 63;
  int lane = threadIdx.x;
  int la = lane & 15;
  int kb = (lane >> 4) << 1;      // fp32 A/B fragment K offset per ISA layout
  int rowbase = b * N_ + I * 16;

  for (int e = lane; e < 16 * C; e += 32) {
    int r = e / C, cc = e % C;
    As[r][cc] = m[(size_t)(rowbase + r) * C + cc];
  }
  if (lane < 16) nA[lane] = nsq[rowbase + lane];

  float td[K_]; int ti[K_];
  for (int k = 0; k < K_; ++k) { td[k] = FLT_MAX; ti[k] = 0; }

  for (int J = 0; J < 64; ++J) {
    __syncthreads();                       // protect Bs/Ds from previous iter
    int colbase = b * N_ + J * 16;
    for (int e = lane; e < 16 * C; e += 32) {
      int r = e / C, cc = e % C;
      Bs[r][cc] = m[(size_t)(colbase + r) * C + cc];
    }
    if (lane < 16) nB[lane] = nsq[colbase + lane];
    __syncthreads();

    v8f acc = {0.f, 0.f, 0.f, 0.f, 0.f, 0.f, 0.f, 0.f};
    for (int kk = 0; kk < C; kk += 4) {
      v2f a, bb;
      a.x  = As[la][kk + kb];  a.y  = As[la][kk + kb + 1];
      bb.x = Bs[la][kk + kb];  bb.y = Bs[la][kk + kb + 1];
      acc = wmma_f32_4(a, bb, acc);
    }
    int hi = lane >> 4;
    for (int v = 0; v < 8; ++v) {
      int il = v + (hi << 3);              // D row for this lane/vgpr
      Ds[il][la] = nA[il] + nB[la] - 2.0f * acc[v];
    }
    __syncthreads();

    if (lane < 16) {                       // stable insertion -> jax tie order
      for (int j2 = 0; j2 < 16; ++j2) {
        float d = Ds[lane][j2];
        if (d < td[K_ - 1]) {
          int p = K_ - 1;
          while (p > 0 && d < td[p - 1]) {
            td[p] = td[p - 1]; ti[p] = ti[p - 1]; --p;
          }
          td[p] = d; ti[p] = J * 16 + j2;
        }
      }
    }
  }
  if (lane < 16) {
    int ro = (rowbase + lane) * K_;
    for (int k = 0; k < K_; ++k) idxO[ro + k] = ti[k];
  }
}

// ---------------------------------------------------------------------------
// Kernel 4: gather 9 neighbors -> LDS, then Y(O x 16) = W(O x C*9) * G(C*9 x 16)
// via fp32 WMMA. Bias preloaded into accumulator. Optional ReLU; final layer
// stores with fused pixel-shuffle.
// ---------------------------------------------------------------------------
template <int CIN, int COUT, int OTILES, bool RELU, bool FINAL>
__global__ __launch_bounds__(256) void gather_gemm_wmma(
    const float* __restrict__ m, const int* __restrict__ idxI,
    const float* __restrict__ wgt, const float* __restrict__ bias,
    float* __restrict__ out) {
  __shared__ float Gs[144][17];            // 16-channel chunk: 16*9=144 K rows
  __shared__ int idxS[16][K_];
  const int CKT = CIN * K_;

  int blk = blockIdx.x;
  int b = blk >> 6;
  int J = blk & 63;
  int tid = threadIdx.x;
  int wave = tid >> 5;
  int lane = tid & 31;
  int la = lane & 15;
  int kb2 = (lane >> 4) << 1;
  int hi = lane >> 4;
  int nbase = b * N_ + J * 16;

  for (int e = tid; e < 16 * K_; e += 256)
    idxS[e / K_][e % K_] = idxI[(size_t)(nbase + e / K_) * K_ + e % K_];

  int orow = wave * 16 + la;               // A-matrix row for this lane
  v8f acc;
  for (int v = 0; v < 8; ++v) {
    int o = wave * 16 + v + (hi << 3);
    acc[v] = (wave < OTILES && o < COUT) ? bias[o] : 0.f;
  }

  for (int cc = 0; cc < CIN; cc += 16) {
    __syncthreads();                       // idxS ready / Gs reads done
    for (int e = tid; e < 144 * 16; e += 256) {
      int t = e & 15;
      int ckl = e >> 4;
      int c = cc + ckl / K_;
      int k = ckl % K_;
      Gs[ckl][t] = m[(size_t)(b * N_ + idxS[t][k]) * CIN + c];
    }
    __syncthreads();

    if (wave < OTILES) {                   // wave-uniform: EXEC all-ones in WMMA
      const float* wrow = wgt + (size_t)orow * CKT + cc * K_;
      bool aval = (orow < COUT);
      for (int kk = 0; kk < 144; kk += 4) {
        v2f a, bb;
        a.x = aval ? wrow[kk + kb2] : 0.f;
        a.y = aval ? wrow[kk + kb2 + 1] : 0.f;
        bb.x = Gs[kk + kb2][la];
        bb.y = Gs[kk + kb2 + 1][la];
        acc = wmma_f32_4(a, bb, acc);
      }
    }
  }

  if (wave < OTILES) {
    for (int v = 0; v < 8; ++v) {
      int o = wave * 16 + v + (hi << 3);
      if (o >= COUT) continue;
      float val = acc[v];
      if (RELU) val = fmaxf(val, 0.f);
      int n = J * 16 + la;
      if (!FINAL) {
        out[(size_t)(b * N_ + n) * COUT + o] = val;
      } else {                             // fused pixel_shuffle(s=2)
        int cch = o >> 2, sy = (o >> 1) & 1, sx = o & 1;
        int h = n >> 5, w2 = n & 31;
        out[((size_t)(b * 3 + cch) * 64 + (2 * h + sy)) * 64 + (2 * w2 + sx)] = val;
      }
    }
  }
}

// ---------------------------------------------------------------------------
extern "C" void kernel_launch(void* const* d_in, const int* in_sizes, int n_in,
                              void* d_out, int out_size, void* d_ws, size_t ws_size,
                              hipStream_t stream) {
  (void)in_sizes; (void)n_in; (void)out_size; (void)ws_size;
  const float* x  = (const float*)d_in[0];
  const float* w1 = (const float*)d_in[1];
  const float* b1 = (const float*)d_in[2];
  const float* w2 = (const float*)d_in[3];
  const float* b2 = (const float*)d_in[4];
  const float* w3 = (const float*)d_in[5];
  const float* b3 = (const float*)d_in[6];
  float* out = (float*)d_out;

  // scratch layout (all 4-byte aligned): m1 | m2 | nsq | idx  (~53 MB)
  float* m1  = (float*)d_ws;                        // 64*1024*64
  float* m2  = m1 + (size_t)B_ * N_ * 64;           // 64*1024*128
  float* nsq = m2 + (size_t)B_ * N_ * 128;          // 64*1024
  int*   idx = (int*)(nsq + (size_t)B_ * N_);       // 64*1024*9

  conv1_relu_tok<<<(B_ * N_ * 64 + 255) / 256, 256, 0, stream>>>(x, w1, b1, m1);

  rownorm<64><<<(B_ * N_ + 255) / 256, 256, 0, stream>>>(m1, nsq);
  topk_wmma<64><<<B_ * 64, 32, 0, stream>>>(m1, nsq, idx);
  gather_gemm_wmma<64, 128, 8, true, false><<<B_ * 64, 256, 0, stream>>>(m1, idx, w2, b2, m2);

  rownorm<128><<<(B_ * N_ + 255) / 256, 256, 0, stream>>>(m2, nsq);
  topk_wmma<128><<<B_ * 64, 32, 0, stream>>>(m2, nsq, idx);
  gather_gemm_wmma<128, 12, 1, false, true><<<B_ * 64, 256, 0, stream>>>(m2, idx, w3, b3, out);
}